// MeanDistanceLoss_78615081386358
// MI455X (gfx1250) — compile-verified
//
#include <hip/hip_runtime.h>
#include <hip/hip_bf16.h>
#include <math.h>

typedef float v2f __attribute__((ext_vector_type(2)));
typedef float v8f __attribute__((ext_vector_type(8)));

#define B_N   256
#define C_N   2048
#define PARTS 64
#define HW    64   // 8*8 spatial

// ---------------- Kernel A: embedding distances to 3 global means + argmin ----------------
__global__ void k_embed_dist(const float* __restrict__ emb,
                             const float* __restrict__ meb,
                             const float* __restrict__ mem,
                             const float* __restrict__ men,
                             float* __restrict__ out)   // out[0..767]=distances_im, out[768..1023]=val
{
    __shared__ float sd0[256], sd1[256], sd2[256];
    const int b = blockIdx.x;
    const int t = threadIdx.x;
    const float* e = emb + (size_t)b * C_N;
    float a0 = 0.f, a1 = 0.f, a2 = 0.f;
    for (int c = t; c < C_N; c += 256) {
        float v = e[c];
        float d0 = v - meb[c]; a0 = fmaf(d0, d0, a0);
        float d1 = v - mem[c]; a1 = fmaf(d1, d1, a1);
        float d2 = v - men[c]; a2 = fmaf(d2, d2, a2);
    }
    sd0[t] = a0; sd1[t] = a1; sd2[t] = a2;
    __syncthreads();
    for (int s = 128; s > 0; s >>= 1) {
        if (t < s) { sd0[t] += sd0[t+s]; sd1[t] += sd1[t+s]; sd2[t] += sd2[t+s]; }
        __syncthreads();
    }
    if (t == 0) {
        float d0 = sqrtf(sd0[0]), d1 = sqrtf(sd1[0]), d2 = sqrtf(sd2[0]);
        out[b*3 + 0] = d0; out[b*3 + 1] = d1; out[b*3 + 2] = d2;
        int v = 0; float m = d0;
        if (d1 < m) { m = d1; v = 1; }
        if (d2 < m) { m = d2; v = 2; }
        out[768 + b] = (float)v;   // argmin, first-min tie-break
    }
}

// ---------------- Kernel B: ||means[s][p]||^2 for s in 0..2, p in 0..63 ----------------
__global__ void k_mean_norm(const float* __restrict__ mb,
                            const float* __restrict__ mm,
                            const float* __restrict__ mn,
                            float* __restrict__ msq)    // msq[s*64+p]
{
    __shared__ float sd[256];
    const int sp = blockIdx.x;                 // 0..191
    const int s = sp >> 6, p = sp & 63;
    const float* m = (s == 0 ? mb : (s == 1 ? mm : mn)) + (size_t)p * C_N;
    const int t = threadIdx.x;
    float a = 0.f;
    for (int c = t; c < C_N; c += 256) { float v = m[c]; a = fmaf(v, v, a); }
    sd[t] = a;
    __syncthreads();
    for (int k = 128; k > 0; k >>= 1) { if (t < k) sd[t] += sd[t+k]; __syncthreads(); }
    if (t == 0) msq[sp] = sd[0];
}

// ---------------- Kernel C: gathered part-embedding distances via f32 WMMA ----------------
// One wave32 per (part p, 16-row b-tile). D = A(16xK) x B(Kx16) accumulated over K=2048
// with V_WMMA_F32_16X16X4_F32. A row m = gathered feature-map channels for batch b0+m;
// B columns 0..2 = the three per-part means (columns 3..15 compute unused garbage).
// Per-lane squared-norm of the gathered row is accumulated in the same pass.
__global__ void k_part_dist_wmma(const float* __restrict__ fm,
                                 const float* __restrict__ mb,
                                 const float* __restrict__ mm,
                                 const float* __restrict__ mn,
                                 const int*   __restrict__ cx,
                                 const int*   __restrict__ cy,
                                 const float* __restrict__ msq,
                                 float* __restrict__ dis)   // dis[b*192 + s*64 + p]
{
    __shared__ float s_norm[32];
    __shared__ float s_dot[16][4];

    const int lane  = threadIdx.x;     // 0..31
    const int m     = lane & 15;       // A-matrix row / B-matrix column for this lane
    const int khalf = lane >> 4;       // which K-pair of the 4-wide K slice
    const int p     = blockIdx.y;      // part
    const int b0    = blockIdx.x << 4; // batch tile base
    const int b     = b0 + m;

    // centroids are (PARTS, B): cell indices in [0,7]
    const int x = cx[p * B_N + b] / 28;
    const int y = cy[p * B_N + b] / 28;
    const float* fp = fm + ((size_t)b * C_N) * HW + (x * 8 + y);

    // B-operand: lane's column n = m; columns >=3 read means_b harmlessly (outputs unused)
    const float* mp = (m == 1 ? mm : (m == 2 ? mn : mb)) + (size_t)p * C_N;

    v8f acc = {0.f, 0.f, 0.f, 0.f, 0.f, 0.f, 0.f, 0.f};
    float nrm = 0.f;

    #pragma unroll 4
    for (int k = 0; k < C_N; k += 4) {
        const int k0 = k + khalf * 2;
        // gathered A elements: stride 256B within a channel column
        float a0 = fp[(size_t)k0 * HW];
        float a1 = fp[(size_t)(k0 + 1) * HW];
        nrm = fmaf(a0, a0, nrm);
        nrm = fmaf(a1, a1, nrm);
        // packed means load (8B aligned: k0 is even)
        float2 mv = *(const float2*)(mp + k0);
        v2f A;  A[0] = a0;    A[1] = a1;
        v2f Bv; Bv[0] = mv.x; Bv[1] = mv.y;
        // (neg_a, A, neg_b, B, c_mod, C, reuse_a, reuse_b)
        acc = __builtin_amdgcn_wmma_f32_16x16x4_f32(false, A, false, Bv,
                                                    (short)0, acc, false, false);
    }

    s_norm[lane] = nrm;
    if (m < 3) {
        // D layout: VGPR i of lane L holds element (row = i + 8*(L/16), col = L%16)
        #pragma unroll
        for (int i = 0; i < 8; ++i) s_dot[i + khalf * 8][m] = acc[i];
    }
    __syncthreads();

    if (lane < 16) {
        const float nn = s_norm[lane] + s_norm[lane + 16];   // both K-halves of row `lane`
        #pragma unroll
        for (int s = 0; s < 3; ++s) {
            float v = nn - 2.f * s_dot[lane][s] + msq[s * 64 + p];
            dis[(size_t)(b0 + lane) * 192 + s * 64 + p] = sqrtf(fmaxf(v, 0.f));
        }
    }
}

// ---------------- Kernel D: hinge-loss reduction ----------------
__global__ void k_loss(const float* __restrict__ dis,
                       const int*  __restrict__ labels,
                       float* __restrict__ out)   // out[1024] = loss
{
    __shared__ float s1[1024];
    __shared__ float s2[1024];
    __shared__ float s_dm;
    const int t = threadIdx.x;

    float sum_dis = 0.f, sum_same = 0.f;
    for (int i = t; i < B_N * 3 * PARTS; i += 1024) sum_dis += dis[i];
    for (int j = t; j < B_N * PARTS; j += 1024) {
        const int b = j >> 6, pp = j & 63;
        sum_same += dis[(size_t)b * 192 + labels[b] * 64 + pp];
    }
    s1[t] = sum_dis; s2[t] = sum_same;
    __syncthreads();
    for (int k = 512; k > 0; k >>= 1) {
        if (t < k) { s1[t] += s1[t+k]; s2[t] += s2[t+k]; }
        __syncthreads();
    }
    if (t == 0) s_dm = (s1[0] - s2[0]) / (float)(B_N * 2 * PARTS);
    __syncthreads();
    const float dm = s_dm;

    float h = 0.f;
    for (int j = t; j < B_N * PARTS; j += 1024) {
        const int b = j >> 6, pp = j & 63;
        const float same = dis[(size_t)b * 192 + labels[b] * 64 + pp];
        h += fmaxf(0.f, 1.f + same - dm);
    }
    s1[t] = h;
    __syncthreads();
    for (int k = 512; k > 0; k >>= 1) { if (t < k) s1[t] += s1[t+k]; __syncthreads(); }
    if (t == 0) out[1024] = s1[0] / (float)(B_N * PARTS);
}

extern "C" void kernel_launch(void* const* d_in, const int* in_sizes, int n_in,
                              void* d_out, int out_size, void* d_ws, size_t ws_size,
                              hipStream_t stream)
{
    const int*   labels = (const int*)  d_in[0];
    const float* emb    = (const float*)d_in[1];
    const float* fm     = (const float*)d_in[2];
    const float* mb     = (const float*)d_in[3];
    const float* mm     = (const float*)d_in[4];
    const float* mn     = (const float*)d_in[5];
    const int*   cx     = (const int*)  d_in[6];
    const int*   cy     = (const int*)  d_in[7];
    const float* meb    = (const float*)d_in[8];
    const float* mem    = (const float*)d_in[9];
    const float* men    = (const float*)d_in[10];
    float* out = (float*)d_out;

    float* msq = (float*)d_ws;   // 192 floats (padded to 256)
    float* dis = msq + 256;      // 49152 floats

    k_embed_dist<<<B_N, 256, 0, stream>>>(emb, meb, mem, men, out);
    k_mean_norm<<<3 * PARTS, 256, 0, stream>>>(mb, mm, mn, msq);
    k_part_dist_wmma<<<dim3(B_N / 16, PARTS), 32, 0, stream>>>(fm, mb, mm, mn, cx, cy, msq, dis);
    k_loss<<<1, 1024, 0, stream>>>(dis, labels, out);
}